// UnifiedSGSv2_9612136808523
// MI455X (gfx1250) — compile-verified
//
#include <hip/hip_runtime.h>
#include <math.h>

// UnifiedSGSv2 for MI455X (gfx1250, wave32).
// VALU-bound stencil workload (25MB tensor is L2-resident): async global->LDS
// halo-tile gathers (ASYNCcnt path), LDS-resident Huber-TV with view-paired
// float2 lanes, hoisted neighbor indices, register-resident own-cell state,
// TRANS-pipe v_rcp for the Huber denominators, exact radix-select medians,
// two-pass stats/output.

#define USE_ASYNC 1

namespace sgs {

constexpr int  Bn    = 32;
constexpr int  Cc    = 3;
constexpr int  Hh    = 256;
constexpr int  Ww    = 256;
constexpr int  Kk    = 5;
constexpr int  NPIX  = Hh * Ww;            // 65536
constexpr int  PLANE = Cc * NPIX;          // 196608 per batch
constexpr long long TOTAL = (long long)Bn * PLANE;  // 6291456
constexpr int  TILE  = 32;
constexpr int  HALO  = 3;                  // 3 TV iterations -> radius 3
constexpr int  LW    = TILE + 2 * HALO;    // 38
constexpr int  LST   = 40;                 // padded LDS row stride (elements)
constexpr int  LSZ   = LST * LW;           // 1520 elements per tile buffer
constexpr int  NEL   = LW * LW;            // 1444 live elements
constexpr int  TAIL  = NEL - 5 * 256;      // 164: threads with a 6th slot

constexpr float EPSf     = 1e-8f;
constexpr float TV_STEPc = 0.2f;

// lambda_k for views 1..4, replicating np.random.RandomState(0) jitter.
__device__ __constant__ float c_lam[4] = {
  0.025244067519636624f,   // i=0
  0.043459911386436829f,   // i=1
  0.059532239387502513f,   // i=2
  0.075673247744953454f    // i=3
};

struct alignas(8) f2 { float x, y; };

__device__ __forceinline__ int imin(int a, int b) { return a < b ? a : b; }
__device__ __forceinline__ int imax(int a, int b) { return a > b ? a : b; }

// 1-ulp hardware reciprocal on the TRANS pipe (co-issues with VALU on CDNA5).
__device__ __forceinline__ float fastrcp(float a) {
#if defined(__has_builtin)
#if __has_builtin(__builtin_amdgcn_rcpf)
  return __builtin_amdgcn_rcpf(a);
#else
  return 1.0f / a;
#endif
#else
  return 1.0f / a;
#endif
}

__device__ __forceinline__ void cswap(float& a, float& b) {
  float lo = fminf(a, b);
  float hi = fmaxf(a, b);
  a = lo; b = hi;
}

// 9-comparator sorting network for 5 elements; returns lower median (s2).
__device__ __forceinline__ float median5(float s0, float s1, float s2,
                                         float s3, float s4) {
  cswap(s0, s1); cswap(s3, s4); cswap(s2, s4); cswap(s2, s3);
  cswap(s1, s4); cswap(s0, s3); cswap(s0, s2); cswap(s1, s3); cswap(s1, s2);
  return s2;
}

__device__ __forceinline__ void wait_async_lds() {
#if USE_ASYNC
#if defined(__has_builtin)
#if __has_builtin(__builtin_amdgcn_s_wait_asynccnt)
  __builtin_amdgcn_s_wait_asynccnt(0);
#else
  asm volatile("s_wait_asynccnt 0x0" ::: "memory");
#endif
#else
  asm volatile("s_wait_asynccnt 0x0" ::: "memory");
#endif
#endif
}

// ---------------------------------------------------------------------------
// Workspace layout (floats):
//   ws[0]      : sum of raw per-pixel MAD over all B*C*H*W
//   ws[1..32]  : per-batch sum of dev over (K,C,H,W)
//   ws[33]     : mad_mean (written by finalize)
//   ws[34..65] : per-batch scale = lower_median(|x|) + EPS
// ---------------------------------------------------------------------------

__global__ void init_ws(float* __restrict__ ws) {
  int t = threadIdx.x;
  if (t < 34) ws[t] = 0.0f;
}

// Exact lower median of |x| per batch via 4-pass MSB->LSB radix-256 select.
// |x| >= 0, so IEEE bit patterns order identically to values.
__global__ __launch_bounds__(1024)
void median_kernel(const float* __restrict__ x, float* __restrict__ ws) {
  const int b = blockIdx.x;
  const float* xb = x + (size_t)b * PLANE;
  __shared__ unsigned hist[256];
  __shared__ unsigned s_prefix, s_rank;
  const unsigned tid = threadIdx.x;
  if (tid == 0) { s_prefix = 0u; s_rank = (unsigned)((PLANE - 1) / 2); } // 98303

  for (int pass = 3; pass >= 0; --pass) {
    if (tid < 256) hist[tid] = 0u;
    __syncthreads();
    const unsigned prefix = s_prefix;
    const unsigned shift  = (unsigned)pass * 8u;
    const unsigned pmask  = (pass == 3) ? 0u : (0xFFFFFFFFu << ((pass + 1) * 8));
    for (int i = (int)tid; i < PLANE; i += 1024) {
      unsigned v = __float_as_uint(fabsf(xb[i]));
      if ((v & pmask) == prefix) {
        unsigned d = (v >> shift) & 0xFFu;
        atomicAdd(&hist[d], 1u);
      }
    }
    __syncthreads();
    if (tid == 0) {
      unsigned r = s_rank, cum = 0u, d = 0u;
      for (d = 0u; d < 256u; ++d) {
        unsigned h = hist[d];
        if (cum + h > r) break;
        cum += h;
      }
      s_prefix = prefix | (d << shift);
      s_rank   = r - cum;
    }
    __syncthreads();
  }
  if (tid == 0) ws[34 + b] = __uint_as_float(s_prefix) + EPSf;
}

// ---------------------------------------------------------------------------
// Tile kernel: one 32x32 interior tile (+3 halo) per block, 256 threads.
// Views {1,2} and {3,4} run as .x/.y lanes of 8-byte LDS elements.
// Clamp-to-edge halo loads exactly reproduce the zero-padded difference
// semantics at image borders (replicated rows/cols => dx=dy=0 there).
// ---------------------------------------------------------------------------
template <bool WRITE_OUT>
__global__ __launch_bounds__(256)
void tile_kernel(const float* __restrict__ x, float* __restrict__ out,
                 float* __restrict__ ws) {
  __shared__ float s_xn[LSZ];  // raw x tile (staging + view-0 extraction)
  __shared__ f2    s_xd[LSZ];  // paired TV iterate
  __shared__ f2    s_gx[LSZ];  // paired gx (reused as reduction scratch)
  __shared__ f2    s_gy[LSZ];  // paired gy

  const int tid  = threadIdx.x;
  const int blk  = blockIdx.x;
  const int tIdx = blk & 63;          // tile index within plane (8x8 tiles)
  const int bc   = blk >> 6;          // plane index = b*Cc + ch
  const int b    = bc / Cc;
  const int ty0  = (tIdx >> 3) * TILE;
  const int tx0  = (tIdx & 7) * TILE;
  const float* plane = x + (size_t)bc * NPIX;

  const float scale     = ws[34 + b];
  const float inv_scale = 1.0f / scale;
  float mad_floor = 0.0f;
  if constexpr (WRITE_OUT) mad_floor = 0.1f * ws[33];

  // ---- hoisted per-thread slot indices (clamped neighbors), computed once --
  int idxC[6], idxR[6], idxD[6], idxL[6], idxU[6];
#pragma unroll
  for (int j = 0; j < 6; ++j) {
    int i = tid + 256 * j;
    if (i >= NEL) i = NEL - 1;          // dummy (guarded at use sites)
    const int r  = i / LW;
    const int c  = i - r * LW;
    const int ic = r * LST + c;
    idxC[j] = ic;
    idxR[j] = ic + ((c < LW - 1) ? 1 : 0);
    idxD[j] = ic + ((r < LW - 1) ? LST : 0);
    idxL[j] = ic - ((c > 0) ? 1 : 0);
    idxU[j] = ic - ((r > 0) ? LST : 0);
  }
  const bool has6 = (tid < TAIL);

  // ---- load 38x38 clamped tile into s_xn (async global -> LDS gather) ----
  const unsigned lds_base = (unsigned)(uintptr_t)(&s_xn[0]);
  for (int i = tid; i < NEL; i += 256) {
    const int r  = i / LW, c2 = i - (i / LW) * LW;
    const int gr = imin(imax(ty0 - HALO + r, 0), Hh - 1);
    const int gc = imin(imax(tx0 - HALO + c2, 0), Ww - 1);
    const float* gp = plane + gr * Ww + gc;
#if USE_ASYNC
    const unsigned ldst = lds_base + (unsigned)((r * LST + c2) * 4);
    const unsigned long long ga = (unsigned long long)(uintptr_t)gp;
    asm volatile("global_load_async_to_lds_b32 %0, %1, off"
                 :: "v"(ldst), "v"(ga) : "memory");
#else
    s_xn[r * LST + c2] = *gp;
#endif
  }
  wait_async_lds();
  __syncthreads();

  // Thread -> 4 interior pixels: rows tyb, tyb+8, tyb+16, tyb+24; col tx.
  const int tx  = tid & 31;
  const int tyb = tid >> 5;

  float v[Kk][4];
#pragma unroll
  for (int j = 0; j < 4; ++j) {
    const int r = tyb + 8 * j;
    v[0][j] = s_xn[(r + HALO) * LST + (tx + HALO)];  // raw x (view 0)
  }

  // normalized input at owned slots, kept in registers for all pairs/iters
  float xnv[6];
#pragma unroll
  for (int j = 0; j < 6; ++j)
    xnv[j] = s_xn[idxC[j]] * inv_scale;

  // ---- 2 view-pairs x 3 Huber-TV iterations, entirely in LDS/registers ----
#pragma unroll
  for (int p = 0; p < 2; ++p) {
    const float lamx = c_lam[2 * p + 0];
    const float lamy = c_lam[2 * p + 1];
    const float dlx  = 0.01f * sqrtf(lamx * 19.999996f);  // lam/(0.05+1e-8)
    const float dly  = 0.01f * sqrtf(lamy * 19.999996f);

    f2 xc[6];
#pragma unroll
    for (int j = 0; j < 6; ++j) {
      if (j < 5 || has6) {
        xc[j].x = xnv[j]; xc[j].y = xnv[j];
        s_xd[idxC[j]] = xc[j];
      }
    }
    __syncthreads();

    for (int it = 0; it < 3; ++it) {
      f2 gxo[6], gyo[6];
      // sweep 1: Huber gradients (v_rcp on TRANS pipe)
#pragma unroll
      for (int j = 0; j < 6; ++j) {
        if (j < 5 || has6) {
          const f2 xr = s_xd[idxR[j]];
          const f2 xd = s_xd[idxD[j]];
          const float dxx = xr.x - xc[j].x, dxy = xr.y - xc[j].y;
          const float dyx = xd.x - xc[j].x, dyy = xd.y - xc[j].y;
          f2 gx, gy;
          gx.x = dxx * fastrcp(dlx + fabsf(dxx));
          gx.y = dxy * fastrcp(dly + fabsf(dxy));
          gy.x = dyx * fastrcp(dlx + fabsf(dyx));
          gy.y = dyy * fastrcp(dly + fabsf(dyy));
          s_gx[idxC[j]] = gx;
          s_gy[idxC[j]] = gy;
          gxo[j] = gx; gyo[j] = gy;
        }
      }
      __syncthreads();
      // sweep 2: divergence + update (own cell state stays in registers)
#pragma unroll
      for (int j = 0; j < 6; ++j) {
        if (j < 5 || has6) {
          const f2 gxl = s_gx[idxL[j]];
          const f2 gyu = s_gy[idxU[j]];
          const float divx = (gxo[j].x - gxl.x) + (gyo[j].x - gyu.x);
          const float divy = (gxo[j].y - gxl.y) + (gyo[j].y - gyu.y);
          f2 t;
          t.x = xc[j].x - TV_STEPc * (xc[j].x - xnv[j] - lamx * divx);
          t.y = xc[j].y - TV_STEPc * (xc[j].y - xnv[j] - lamy * divy);
          xc[j] = t;
          s_xd[idxC[j]] = t;
        }
      }
      __syncthreads();
    }

#pragma unroll
    for (int j = 0; j < 4; ++j) {
      const int r = tyb + 8 * j;
      const f2 t = s_xd[(r + HALO) * LST + (tx + HALO)];
      v[1 + 2 * p][j] = t.x * scale;
      v[2 + 2 * p][j] = t.y * scale;
    }
    __syncthreads();  // before next pair overwrites s_xd
  }

  // ---- per-pixel 5-way median / MAD fusion ----
  float local_mad = 0.0f, local_dev = 0.0f;
#pragma unroll
  for (int j = 0; j < 4; ++j) {
    const float a0 = v[0][j], a1 = v[1][j], a2 = v[2][j],
                a3 = v[3][j], a4 = v[4][j];
    const float med = median5(a0, a1, a2, a3, a4);
    const float d0 = fabsf(a0 - med), d1 = fabsf(a1 - med),
                d2 = fabsf(a2 - med), d3 = fabsf(a3 - med),
                d4 = fabsf(a4 - med);
    const float mad_raw = median5(d0, d1, d2, d3, d4);
    if constexpr (WRITE_OUT) {
      const float mad = fmaxf(mad_raw, mad_floor);
      const float inv = 0.5f * fastrcp(mad);  // 1/(FUSION_SCALE*mad), SCALE=2
      const float w0 = __expf(-d0 * inv), w1 = __expf(-d1 * inv),
                  w2 = __expf(-d2 * inv), w3 = __expf(-d3 * inv),
                  w4 = __expf(-d4 * inv);
      const float num = w0 * a0 + w1 * a1 + w2 * a2 + w3 * a3 + w4 * a4;
      const float den = w0 + w1 + w2 + w3 + w4 + EPSf;
      const int grow = ty0 + tyb + 8 * j;
      const int gcol = tx0 + tx;
      out[(size_t)bc * NPIX + grow * Ww + gcol] = num / den;
    } else {
      local_mad += mad_raw;
      local_dev += d0 + d1 + d2 + d3 + d4;
    }
  }

  if constexpr (!WRITE_OUT) {
    float* rmad = (float*)s_gx;
    float* rdev = (float*)s_gy;
    rmad[tid] = local_mad;
    rdev[tid] = local_dev;
    __syncthreads();
    for (int off = 128; off > 0; off >>= 1) {
      if (tid < off) {
        rmad[tid] += rmad[tid + off];
        rdev[tid] += rdev[tid + off];
      }
      __syncthreads();
    }
    if (tid == 0) {
      atomicAdd(&ws[0], rmad[0]);
      atomicAdd(&ws[1 + b], rdev[0]);
    }
  }
}

__global__ void finalize_kernel(float* __restrict__ ws, float* __restrict__ out) {
  const int t = threadIdx.x;
  if (t == 0) ws[33] = ws[0] / (float)TOTAL;  // mad.mean()
  if (t < Bn) {
    const float D = ws[1 + t] / (float)((long long)Kk * PLANE);  // 983040
    const float z = (D - 0.02f) / 0.01f;
    out[TOTAL + t] = 1.0f / (1.0f + expf(-z));  // beta in [0,1]
  }
}

}  // namespace sgs

extern "C" void kernel_launch(void* const* d_in, const int* in_sizes, int n_in,
                              void* d_out, int out_size, void* d_ws, size_t ws_size,
                              hipStream_t stream) {
  using namespace sgs;
  const float* x = (const float*)d_in[0];
  float* out = (float*)d_out;
  float* ws  = (float*)d_ws;

  init_ws<<<1, 64, 0, stream>>>(ws);
  median_kernel<<<Bn, 1024, 0, stream>>>(x, ws);
  tile_kernel<false><<<Bn * Cc * 64, 256, 0, stream>>>(x, out, ws);
  finalize_kernel<<<1, 64, 0, stream>>>(ws, out);
  tile_kernel<true><<<Bn * Cc * 64, 256, 0, stream>>>(x, out, ws);
}